// SpMiddleFHD_59347858096779
// MI455X (gfx1250) — compile-verified
//
#include <hip/hip_runtime.h>
#include <hip/hip_bf16.h>
#include <cstdint>

// ---------------- problem constants (fixed by the reference) ----------------
namespace {
constexpr int       kN     = 40000;          // B * N_PER voxels
constexpr int       kC1    = 128;            // input channels
constexpr int       kB     = 2;
constexpr int       kDz = 41, kHy = 400, kWx = 352;
constexpr int       kGD = kDz + 2, kGH = kHy + 2, kGW = kWx + 2;   // padded grid
constexpr int       kDo = 21, kHo = 200, kWo = 176;                // output dims
constexpr long long kGridN  = (long long)kB * kGD * kGH * kGW;     // 12,238,488
constexpr long long kOutVox = (long long)kB * kDo * kHo * kWo;     // 1,478,400
constexpr long long kOutN   = kOutVox * 32;                        // 47,308,800
constexpr int       kK1 = 27 * kC1;          // 3456  (layer-1 GEMM K)
constexpr int       kK2 = 27 * 16;           // 432   (layer-2 GEMM K)
constexpr int       kTiles = kN / 16;        // 2500 exactly
}

typedef __attribute__((ext_vector_type(2))) float v2f;
typedef __attribute__((ext_vector_type(8))) float v8f;

// ---------------- helpers ----------------
__global__ void fill_i32_kernel(int* __restrict__ p, long long n, int v) {
  long long i = (long long)blockIdx.x * blockDim.x + threadIdx.x;
  const long long stride = (long long)gridDim.x * blockDim.x;
  for (; i < n; i += stride) p[i] = v;
}

__global__ void fill_f32_kernel(float* __restrict__ p, long long n, float v) {
  long long i = (long long)blockIdx.x * blockDim.x + threadIdx.x;
  const long long stride = (long long)gridDim.x * blockDim.x;
  for (; i < n; i += stride) p[i] = v;
}

__global__ void relu_kernel(float* __restrict__ p, long long n) {
  long long i = (long long)blockIdx.x * blockDim.x + threadIdx.x;
  const long long stride = (long long)gridDim.x * blockDim.x;
  for (; i < n; i += stride) p[i] = fmaxf(p[i], 0.0f);
}

// grid[b, z+1, y+1, x+1] = n
__global__ void scatter_grid_kernel(const int* __restrict__ coors, int* __restrict__ grid) {
  int n = blockIdx.x * blockDim.x + threadIdx.x;
  if (n >= kN) return;
  const int b = coors[n * 4 + 0], z = coors[n * 4 + 1];
  const int y = coors[n * 4 + 2], x = coors[n * 4 + 3];
  const long long gi = (((long long)b * kGD + (z + 1)) * kGH + (y + 1)) * kGW + (x + 1);
  grid[gi] = n;
}

// nbr[n*27 + k] = grid[b, z+dz, y+dy, x+dx], dz/dy/dx in 0..2
__global__ void build_nbr_kernel(const int* __restrict__ coors, const int* __restrict__ grid,
                                 int* __restrict__ nbr) {
  int t = blockIdx.x * blockDim.x + threadIdx.x;
  if (t >= kN * 27) return;
  const int n = t / 27, k = t % 27;
  const int dz = k / 9, dy = (k / 3) % 3, dx = k % 3;
  const int b = coors[n * 4 + 0], z = coors[n * 4 + 1];
  const int y = coors[n * 4 + 2], x = coors[n * 4 + 3];
  const long long gi = (((long long)b * kGD + (z + dz)) * kGH + (y + dy)) * kGW + (x + dx);
  nbr[t] = grid[gi];
}

// W (flat [k][n], n<16) -> Wt [n][k]  : makes WMMA B-fragment loads contiguous float2
__global__ void repack_w_kernel(const float* __restrict__ W, float* __restrict__ Wt, int K) {
  int t = blockIdx.x * blockDim.x + threadIdx.x;
  if (t >= K * 16) return;
  const int k = t >> 4, n = t & 15;
  Wt[(size_t)n * K + k] = W[t];
}

// ---------------- submanifold conv via V_WMMA_F32_16X16X4_F32 ----------------
// One wave handles a 16-voxel tile (M=16), produces all 16 out channels (N=16),
// K = 27 * CIN, stepped 4 at a time through the fp32 WMMA.
// f32 A 16x4 layout: lane = M + 16*(K>>1), vgpr = K&1  (ISA 7.12.2)
// f32 B 4x16 layout (mirrored): lane = N + 16*(K>>1), vgpr = K&1
// f32 C/D 16x16:  lane = N + 16*(M>>3), vgpr = M&7
// Invalid neighbors read a zero-filled dummy row instead of per-fragment
// cndmask selects: one pointer select per ko, zero VALU in the K inner loop.
template <int CIN, int KTOT>
__global__ __launch_bounds__(128)
void subm_kernel(const float* __restrict__ feat, const float* __restrict__ Wt,
                 const int* __restrict__ nbr, const float* __restrict__ zrow,
                 float* __restrict__ xout) {
  const int lane = threadIdx.x & 31;
  const int tile = blockIdx.x * 4 + (threadIdx.x >> 5);  // 625 blocks * 4 waves = 2500 tiles
  const int nn = lane & 15;   // A row (M) for gather, B/D column (N)
  const int h  = lane >> 4;   // which K pair within a K-step of 4
  const int vox = tile * 16 + nn;

  v8f acc = {};

  for (int ko = 0; ko < 27; ++ko) {
    const int idx = nbr[vox * 27 + ko];
    // invalid -> zero row (always in-bounds, values are 0.0f); no per-load select
    const float* rowp = (idx >= 0 ? feat + (size_t)idx * CIN : zrow) + 2 * h;
    const float* wp   = Wt + (size_t)nn * KTOT + ko * CIN + 2 * h;
#pragma unroll
    for (int cb = 0; cb < CIN; cb += 4) {
      const v2f a     = *(const v2f*)(rowp + cb);
      const v2f bfrag = *(const v2f*)(wp + cb);
      acc = __builtin_amdgcn_wmma_f32_16x16x4_f32(
          /*neg_a=*/false, a, /*neg_b=*/false, bfrag,
          /*c_mod=*/(short)0, acc, /*reuse_a=*/false, /*reuse_b=*/false);
    }
  }

  float* orow = xout + (size_t)tile * 16 * 16;
#pragma unroll
  for (int r = 0; r < 8; ++r) {
    const int m = r + 8 * h;
    orow[m * 16 + nn] = fmaxf(acc[r], 0.0f);      // fused ReLU
  }
}

// ---------------- layer 3: 16->32 per (voxel, offset), scatter-add ----------------
__global__ void scatter_conv3_kernel(const float* __restrict__ x2, const int* __restrict__ coors,
                                     const float* __restrict__ W3, float* __restrict__ out) {
  int t = blockIdx.x * blockDim.x + threadIdx.x;
  if (t >= kN * 27) return;
  const int n = t / 27, k = t % 27;
  const int dz = k / 9, dy = (k / 3) % 3, dx = k % 3;
  const int b = coors[n * 4 + 0], z = coors[n * 4 + 1];
  const int y = coors[n * 4 + 2], x = coors[n * 4 + 3];
  const int oz = z + 1 - dz, oy = y + 1 - dy, ox = x + 1 - dx;
  if ((oz | oy | ox) & 1) return;                 // all must be even (oz >= -1 so sign-safe)
  const int pz = oz >> 1, py = oy >> 1, px = ox >> 1;
  if (pz < 0 || pz >= kDo || py < 0 || py >= kHo || px < 0 || px >= kWo) return;
  const long long flat = (((long long)b * kDo + pz) * kHo + py) * kWo + px;

  float xi[16];
#pragma unroll
  for (int c = 0; c < 16; ++c) xi[c] = x2[(size_t)n * 16 + c];
  const float* w = W3 + (size_t)k * 16 * 32;      // W3 reshape(27,16,32)
  float* op = out + flat * 32;
#pragma unroll
  for (int d = 0; d < 32; ++d) {
    float s = 0.0f;
#pragma unroll
    for (int c = 0; c < 16; ++c) s = fmaf(xi[c], w[c * 32 + d], s);
    __hip_atomic_fetch_add(op + d, s, __ATOMIC_RELAXED, __HIP_MEMORY_SCOPE_AGENT);
  }
}

// ---------------- launch ----------------
extern "C" void kernel_launch(void* const* d_in, const int* in_sizes, int n_in,
                              void* d_out, int out_size, void* d_ws, size_t ws_size,
                              hipStream_t stream) {
  (void)in_sizes; (void)n_in; (void)out_size; (void)ws_size;
  const float* feat  = (const float*)d_in[0];   // (40000,128) f32
  const int*   coors = (const int*)d_in[1];     // (40000,4)   i32
  const float* W1    = (const float*)d_in[2];   // (3,3,3,128,16)
  const float* W2    = (const float*)d_in[3];   // (3,3,3,16,16)
  const float* W3    = (const float*)d_in[4];   // (3,3,3,16,32)
  float* out = (float*)d_out;                   // (2,21,200,176,32)

  // workspace carve-out (~59 MB)
  char* ws = (char*)d_ws;
  size_t off = 0;
  auto carve = [&](size_t bytes) -> void* {
    void* p = ws + off;
    off = (off + bytes + 255) & ~(size_t)255;
    return p;
  };
  int*   grid = (int*)carve((size_t)kGridN * 4);
  int*   nbr  = (int*)carve((size_t)kN * 27 * 4);
  float* x1   = (float*)carve((size_t)kN * 16 * 4);
  float* x2   = (float*)carve((size_t)kN * 16 * 4);
  float* W1t  = (float*)carve((size_t)kK1 * 16 * 4);
  float* W2t  = (float*)carve((size_t)kK2 * 16 * 4);
  float* zrow = (float*)carve((size_t)kC1 * 4);   // zero dummy feature row

  fill_i32_kernel<<<2048, 256, 0, stream>>>(grid, kGridN, -1);
  fill_f32_kernel<<<1, 128, 0, stream>>>(zrow, kC1, 0.0f);
  scatter_grid_kernel<<<(kN + 255) / 256, 256, 0, stream>>>(coors, grid);
  build_nbr_kernel<<<(kN * 27 + 255) / 256, 256, 0, stream>>>(coors, grid, nbr);
  repack_w_kernel<<<(kK1 * 16 + 255) / 256, 256, 0, stream>>>(W1, W1t, kK1);
  repack_w_kernel<<<(kK2 * 16 + 255) / 256, 256, 0, stream>>>(W2, W2t, kK2);

  subm_kernel<128, kK1><<<kTiles / 4, 128, 0, stream>>>(feat, W1t, nbr, zrow, x1);
  subm_kernel<16,  kK2><<<kTiles / 4, 128, 0, stream>>>(x1,   W2t, nbr, zrow, x2);

  fill_f32_kernel<<<4096, 256, 0, stream>>>(out, kOutN, 0.0f);
  scatter_conv3_kernel<<<(kN * 27 + 255) / 256, 256, 0, stream>>>(x2, coors, W3, out);
  relu_kernel<<<4096, 256, 0, stream>>>(out, kOutN);
}